// LIPooling_71975061946917
// MI455X (gfx1250) — compile-verified
//
#include <hip/hip_runtime.h>
#include <stdint.h>

// LIPooling: out = max_p( sumspike_prev[...,p,...] + unfold2x2(x)[p] )
// B=32, C=128, H=W=64, oh=ow=32, K=S=2. Pure streaming op (~144 MiB @ 23.3 TB/s).
// Strategy: TDM async-load each 16KB x-plane into LDS (gfx1250 tensor path),
// volatile addrspace(3) b128 LDS reads (ds_load_b128; TDM writes are invisible
// to alias analysis so plain loads would fold to NaN), b128 coalesced global
// loads for sumspike, VALU add/max, b128 stores.

#define B_  32
#define C_  128
#define H_  64
#define W_  64
#define OH_ 32
#define OW_ 32

typedef __attribute__((ext_vector_type(4))) unsigned int v4u;
typedef __attribute__((ext_vector_type(8))) int          v8i;
typedef __attribute__((ext_vector_type(4))) int          v4i;
typedef __attribute__((ext_vector_type(4))) float        v4f;

// Volatile load of 16 bytes straight from LDS address space -> ds_load_b128.
typedef __attribute__((address_space(3))) const volatile v4f* lds_cv4f_p;
__device__ __forceinline__ v4f lds_load4(unsigned byte_off) {
    return *((lds_cv4f_p)byte_off);
}

__global__ __launch_bounds__(256) void lipool_kernel(const float* __restrict__ x,
                                                     const float* __restrict__ ss,
                                                     float* __restrict__ out)
{
    // One (b,c) plane per workgroup.
    __shared__ float xs[H_ * W_];   // 16 KB

    // LDS byte offset of xs (low 32 bits of the flat LDS address are the
    // offset). Also keeps the LDS allocation alive.
    const unsigned xbase = (unsigned)(size_t)&xs[0];

    const int plane = blockIdx.x;                       // b*C + c
    const float* xp = x + (size_t)plane * (H_ * W_);

    // ---- TDM: async DMA the whole 64x64 f32 plane into LDS (wave 0 only) ----
    if (threadIdx.x < 32) {
        const uint64_t ga = (uint64_t)(uintptr_t)xp;

        // D# group 0 (128b): count=1 | lds_addr | global_addr | type=2
        v4u g0;
        g0[0] = 1u;                                      // count=1, user mode
        g0[1] = xbase;                                   // lds_addr (bytes)
        g0[2] = (unsigned)(ga & 0xFFFFFFFFu);            // global_addr[31:0]
        g0[3] = (unsigned)(((ga >> 32) & 0x01FFFFFFu)    // global_addr[56:32]
                           | 0x80000000u);               // type=2 (bits 127:126)

        // D# group 1 (256b): data_size=4B, tensor 64x64, tile 64x64, stride0=64
        v8i g1;
        g1[0] = (int)(2u << 16);        // data_size=2 (4 bytes); wg_mask=0
        g1[1] = (int)(64u << 16);       // tensor_dim0[15:0] @ bits 63:48
        g1[2] = (int)(64u << 16);       // tensor_dim1[15:0] @ bits 111:80 (low16)
        g1[3] = (int)(64u << 16);       // tile_dim0 @ bits 127:112
        g1[4] = 64;                     // tile_dim1 @ bits 143:128 (tile_dim2=0)
        g1[5] = 64;                     // tensor_dim0_stride[31:0]
        g1[6] = 0;                      // stride0 hi / tensor_dim1_stride (unused, 2D)
        g1[7] = 0;

        v4i gz4 = {0, 0, 0, 0};                    // groups 2/3: zero (2-D tensor)
        v8i gz8 = {0, 0, 0, 0, 0, 0, 0, 0};        // extra group operand (clang-23 form)

        __builtin_amdgcn_tensor_load_to_lds(g0, g1, gz4, gz4, gz8, 0);
        __builtin_amdgcn_s_wait_tensorcnt(0);            // wave0's TDM complete
    }
    __syncthreads();                                     // publish LDS to all waves

    // ---- compute: each thread emits 4 outputs (row i, cols j..j+3) ----
    const int t  = threadIdx.x;
    const int i  = t >> 3;               // 0..31 output row
    const int j  = (t & 7) << 2;         // 0,4,...,28 output col (float4)

    // x rows 2i and 2i+1, cols 2j..2j+7: four ds_load_b128 from LDS.
    const unsigned r0 = xbase + (unsigned)(((2 * i)     * W_ + 2 * j) * 4);
    const unsigned r1 = xbase + (unsigned)(((2 * i + 1) * W_ + 2 * j) * 4);
    const v4f a = lds_load4(r0);
    const v4f b = lds_load4(r0 + 16);
    const v4f c = lds_load4(r1);
    const v4f d = lds_load4(r1 + 16);

    // sumspike planes p=0..3, each (oh*ow)=1024 apart; fully coalesced b128
    const size_t ssb = (size_t)plane * (4 * OH_ * OW_) + (size_t)i * OW_ + j;
    const v4f s0 = *(const v4f*)&ss[ssb + 0 * OH_ * OW_];
    const v4f s1 = *(const v4f*)&ss[ssb + 1 * OH_ * OW_];
    const v4f s2 = *(const v4f*)&ss[ssb + 2 * OH_ * OW_];
    const v4f s3 = *(const v4f*)&ss[ssb + 3 * OH_ * OW_];

    // p0 -> x[2i][2j], p1 -> x[2i][2j+1], p2 -> x[2i+1][2j], p3 -> x[2i+1][2j+1]
    v4f o;
    o.x = fmaxf(fmaxf(s0.x + a.x, s1.x + a.y), fmaxf(s2.x + c.x, s3.x + c.y));
    o.y = fmaxf(fmaxf(s0.y + a.z, s1.y + a.w), fmaxf(s2.y + c.z, s3.y + c.w));
    o.z = fmaxf(fmaxf(s0.z + b.x, s1.z + b.y), fmaxf(s2.z + d.x, s3.z + d.y));
    o.w = fmaxf(fmaxf(s0.w + b.z, s1.w + b.w), fmaxf(s2.w + d.z, s3.w + d.w));

    *(v4f*)&out[(size_t)plane * (OH_ * OW_) + (size_t)i * OW_ + j] = o;
}

extern "C" void kernel_launch(void* const* d_in, const int* in_sizes, int n_in,
                              void* d_out, int out_size, void* d_ws, size_t ws_size,
                              hipStream_t stream)
{
    (void)in_sizes; (void)n_in; (void)out_size; (void)d_ws; (void)ws_size;
    const float* x  = (const float*)d_in[0];   // (32,128,64,64)  f32
    const float* ss = (const float*)d_in[1];   // (32,128,4,32,32) f32
    float* out      = (float*)d_out;           // (32,128,32,32)  f32

    lipool_kernel<<<dim3(B_ * C_), dim3(256), 0, stream>>>(x, ss, out);
}